// DecisionActionAuxiliaryHeads_87780541596335
// MI455X (gfx1250) — compile-verified
//
#include <hip/hip_runtime.h>
#include <hip/hip_bf16.h>
#include <math.h>

typedef __attribute__((ext_vector_type(2))) float v2f;
typedef __attribute__((ext_vector_type(8))) float v8f;

#define B_SZ   128
#define S_SZ   512
#define H_SZ   2048
#define P_SZ   2048
#define N_NAMES 64
#define N_ARGS  128
#define C_FULL  512
#define LOGIT_FLOOR -1e9f

// ---------------------------------------------------------------------------
// Stage 1: pooled[b,:] = hidden_states[b, clip(sum(mask),1)-1, :]
// ---------------------------------------------------------------------------
__global__ void pool_kernel(const float* __restrict__ hs,
                            const int* __restrict__ amask,
                            float* __restrict__ pooled) {
  const int b = blockIdx.x;
  const int tid = threadIdx.x;                    // 256 threads
  __shared__ int red[256];
  int s = 0;
  for (int i = tid; i < S_SZ; i += 256) s += amask[b * S_SZ + i];
  red[tid] = s;
  __syncthreads();
  for (int off = 128; off > 0; off >>= 1) {
    if (tid < off) red[tid] += red[tid + off];
    __syncthreads();
  }
  int L = red[0];
  if (L < 1) L = 1;
  L -= 1;
  const float* src = hs + ((size_t)b * S_SZ + (size_t)L) * H_SZ;
  float* dst = pooled + (size_t)b * H_SZ;
  for (int i = tid; i < H_SZ; i += 256) dst[i] = src[i];
}

// ---------------------------------------------------------------------------
// Split-K f32 WMMA GEMM. One wave computes a 32x64 partial tile with
// V_WMMA_F32_16X16X4_F32 over K-chunk kc, writing RAW partial sums:
//   P[kc*kchunk_stride + z*c_zstride + m*ldc + n]
// grid = (N/64, M/32, Z*kchunks)
// ---------------------------------------------------------------------------
__global__ __launch_bounds__(32)
void gemm_wmma_f32(const float* __restrict__ A, int lda,
                   const float* __restrict__ W, int ldw, long long w_zstride,
                   float* __restrict__ P, int ldc, long long c_zstride,
                   long long kchunk_stride, int kchunks, int klen) {
  const int lane  = threadIdx.x;        // 0..31 (wave32)
  const int col16 = lane & 15;
  const int khalf = lane >> 4;          // 0 or 1
  const int m0 = blockIdx.y * 32;
  const int n0 = blockIdx.x * 64;
  const int zt = blockIdx.z;
  const int kc = zt % kchunks;
  const int z  = zt / kchunks;
  const int k0 = kc * klen;

  const float* Wz = W + (size_t)z * (size_t)w_zstride;
  float*       Pz = P + (size_t)kc * (size_t)kchunk_stride +
                        (size_t)z * (size_t)c_zstride;

  v8f accA[4] = {{}, {}, {}, {}};   // rows m0..m0+15
  v8f accB[4] = {{}, {}, {}, {}};   // rows m0+16..m0+31

  // A fragment: lane holds row (m&15)=col16, K = 2*khalf + j
  const float* Arow0 = A + (size_t)(m0 + col16) * (size_t)lda;
  const float* Arow1 = Arow0 + (size_t)16 * (size_t)lda;

#pragma unroll 8
  for (int k = k0; k < k0 + klen; k += 4) {
    const int kr = k + 2 * khalf;
    const v2f a0 = *(const v2f*)(Arow0 + kr);
    const v2f a1 = *(const v2f*)(Arow1 + kr);

    const float* W0 = Wz + (size_t)kr * (size_t)ldw + n0 + col16;
    const float* W1 = W0 + ldw;
    v2f b0, b1, b2, b3;
    b0.x = W0[0];  b0.y = W1[0];
    b1.x = W0[16]; b1.y = W1[16];
    b2.x = W0[32]; b2.y = W1[32];
    b3.x = W0[48]; b3.y = W1[48];

    accA[0] = __builtin_amdgcn_wmma_f32_16x16x4_f32(false, a0, false, b0, (short)0, accA[0], false, false);
    accA[1] = __builtin_amdgcn_wmma_f32_16x16x4_f32(false, a0, false, b1, (short)0, accA[1], false, false);
    accA[2] = __builtin_amdgcn_wmma_f32_16x16x4_f32(false, a0, false, b2, (short)0, accA[2], false, false);
    accA[3] = __builtin_amdgcn_wmma_f32_16x16x4_f32(false, a0, false, b3, (short)0, accA[3], false, false);
    accB[0] = __builtin_amdgcn_wmma_f32_16x16x4_f32(false, a1, false, b0, (short)0, accB[0], false, false);
    accB[1] = __builtin_amdgcn_wmma_f32_16x16x4_f32(false, a1, false, b1, (short)0, accB[1], false, false);
    accB[2] = __builtin_amdgcn_wmma_f32_16x16x4_f32(false, a1, false, b2, (short)0, accB[2], false, false);
    accB[3] = __builtin_amdgcn_wmma_f32_16x16x4_f32(false, a1, false, b3, (short)0, accB[3], false, false);
  }

  // D layout: VGPR r -> row (khalf*8 + r), col = lane&15 within 16-wide tile
  const int mbase = m0 + khalf * 8;
#pragma unroll
  for (int r = 0; r < 8; ++r) {
#pragma unroll
    for (int nt = 0; nt < 4; ++nt) {
      const int n = n0 + nt * 16 + col16;
      Pz[(size_t)(mbase + r) * (size_t)ldc + n]      = accA[nt][r];
      Pz[(size_t)(mbase + 16 + r) * (size_t)ldc + n] = accB[nt][r];
    }
  }
}

// ---------------------------------------------------------------------------
// Reduce split-K partials + fused epilogue (bias, SiLU, adapter row-add).
// out[i] = act( sum_kc P[kc*cstride + i] + bias[i % bias_mod] ) + rowadd
// one element per thread; total elements = gridDim.x * 256 exactly.
// ---------------------------------------------------------------------------
__global__ void reduce_epilogue(const float* __restrict__ P, long long cstride,
                                int nchunks, float* __restrict__ out, int ld,
                                const float* __restrict__ bias, int bias_mod,
                                int act_silu,
                                const float* __restrict__ rowadd_tab,
                                const int* __restrict__ rowadd_ids,
                                const float* __restrict__ scale_ptr,
                                int rowadd_ld) {
  const long long i = (long long)blockIdx.x * blockDim.x + threadIdx.x;
  float s = 0.0f;
  for (int c = 0; c < nchunks; ++c) s += P[(size_t)c * (size_t)cstride + i];
  if (bias != nullptr) s += bias[i % bias_mod];
  if (act_silu) s = s / (1.0f + expf(-s));     // silu
  if (rowadd_tab != nullptr) {
    const long long m = i / ld;
    const long long n = i % ld;
    s += scale_ptr[0] * rowadd_tab[(size_t)rowadd_ids[m] * rowadd_ld + n];
  }
  out[i] = s;
}

// ---------------------------------------------------------------------------
// Losses: name head + arg head (arg logits read from ALL, barg folded in).
// ---------------------------------------------------------------------------
__global__ void loss_kernel(const float* __restrict__ name_logits,
                            const int* __restrict__ name_masks,
                            const int* __restrict__ tgt_name,
                            const float* __restrict__ ALL,
                            const int* __restrict__ arg_masks,
                            const int* __restrict__ tgt_arg,
                            float* __restrict__ out_loss) {
  const int b = threadIdx.x;              // 128 threads
  __shared__ float red[128];

  const float* nl = name_logits + (size_t)b * N_NAMES;
  const int*   nm = name_masks + (size_t)b * N_NAMES;
  bool anyv = false;
  for (int j = 0; j < N_NAMES; ++j) anyv = anyv || (nm[j] != 0);
  float mx = -INFINITY;
  for (int j = 0; j < N_NAMES; ++j) {
    float x = (anyv && nm[j] == 0) ? LOGIT_FLOOR : nl[j];
    mx = fmaxf(mx, x);
  }
  float se = 0.0f;
  for (int j = 0; j < N_NAMES; ++j) {
    float x = (anyv && nm[j] == 0) ? LOGIT_FLOOR : nl[j];
    se += expf(x - mx);
  }
  const int tn = tgt_name[b];
  float xt = (anyv && nm[tn] == 0) ? LOGIT_FLOOR : nl[tn];
  float l_name = -(xt - mx - logf(se));

  const float* al = ALL + (size_t)b * N_NAMES * N_ARGS + (size_t)tn * N_ARGS;
  const int*   am = arg_masks + (size_t)b * N_ARGS;
  bool anya = false;
  for (int j = 0; j < N_ARGS; ++j) anya = anya || (am[j] != 0);
  float mxa = -INFINITY;
  for (int j = 0; j < N_ARGS; ++j) {
    float x = (anya && am[j] == 0) ? LOGIT_FLOOR : al[j];
    mxa = fmaxf(mxa, x);
  }
  float sea = 0.0f;
  for (int j = 0; j < N_ARGS; ++j) {
    float x = (anya && am[j] == 0) ? LOGIT_FLOOR : al[j];
    sea += expf(x - mxa);
  }
  const int ta = tgt_arg[b];
  float xta = (anya && am[ta] == 0) ? LOGIT_FLOOR : al[ta];
  float l_arg = -(xta - mxa - logf(sea));

  red[b] = l_name + l_arg;
  __syncthreads();
  for (int off = 64; off > 0; off >>= 1) {
    if (b < off) red[b] += red[b + off];
    __syncthreads();
  }
  if (b == 0) out_loss[0] = red[0] / (float)B_SZ;
}

// ---------------------------------------------------------------------------
// Full candidate scoring + first-occurrence argmax.
// ---------------------------------------------------------------------------
__global__ void scores_kernel(const float* __restrict__ name_logits,
                              const float* __restrict__ ALL,
                              const int* __restrict__ cn,
                              const int* __restrict__ ca,
                              const int* __restrict__ cmask,
                              float* __restrict__ scores_out,
                              float* __restrict__ preds_out) {
  const int b = blockIdx.x;
  const int tid = threadIdx.x;            // 256 threads
  __shared__ int ired[256];
  __shared__ float vred[256];
  __shared__ int xred[256];

  int la = 0;
  for (int c = tid; c < C_FULL; c += 256) la |= cmask[(size_t)b * C_FULL + c];
  ired[tid] = la;
  __syncthreads();
  for (int off = 128; off > 0; off >>= 1) {
    if (tid < off) ired[tid] |= ired[tid + off];
    __syncthreads();
  }
  const bool anyv = (ired[0] != 0);

  float best = -INFINITY;
  int bestc = C_FULL;
  for (int c = tid; c < C_FULL; c += 256) {
    const int n = cn[(size_t)b * C_FULL + c];
    const int a = ca[(size_t)b * C_FULL + c];
    float sc = name_logits[(size_t)b * N_NAMES + n] +
               ALL[(size_t)b * N_NAMES * N_ARGS + (size_t)n * N_ARGS + a];
    if (anyv && cmask[(size_t)b * C_FULL + c] == 0) sc = LOGIT_FLOOR;
    scores_out[(size_t)b * C_FULL + c] = sc;
    if (sc > best) { best = sc; bestc = c; }   // ascending c => first occurrence
  }
  vred[tid] = best;
  xred[tid] = bestc;
  __syncthreads();
  for (int off = 128; off > 0; off >>= 1) {
    if (tid < off) {
      const float ov = vred[tid + off];
      const int oi = xred[tid + off];
      if (ov > vred[tid] || (ov == vred[tid] && oi < xred[tid])) {
        vred[tid] = ov;
        xred[tid] = oi;
      }
    }
    __syncthreads();
  }
  if (tid == 0) preds_out[b] = (float)xred[0];
}

// ---------------------------------------------------------------------------
extern "C" void kernel_launch(void* const* d_in, const int* in_sizes, int n_in,
                              void* d_out, int out_size, void* d_ws, size_t ws_size,
                              hipStream_t stream) {
  const float* hidden   = (const float*)d_in[0];
  const int*   amask    = (const int*)d_in[1];
  const int*   adap_ids = (const int*)d_in[2];
  const int*   tgt_name = (const int*)d_in[3];
  const int*   tgt_arg  = (const int*)d_in[4];
  const int*   name_cm  = (const int*)d_in[5];
  const int*   arg_cm   = (const int*)d_in[6];
  const int*   full_cm  = (const int*)d_in[7];
  const int*   full_cn  = (const int*)d_in[8];
  const int*   full_ca  = (const int*)d_in[9];
  const float* W1       = (const float*)d_in[10];
  const float* W2       = (const float*)d_in[11];
  const float* adap_emb = (const float*)d_in[12];
  const float* adap_sc  = (const float*)d_in[13];
  const float* Wname    = (const float*)d_in[14];
  const float* bname    = (const float*)d_in[15];
  const float* Warg     = (const float*)d_in[16];
  const float* barg     = (const float*)d_in[17];

  // workspace layout (floats)
  float* ws = (float*)d_ws;
  float* pooled      = ws;                            // 128*2048
  float* T1          = pooled + B_SZ * P_SZ;          // 128*2048
  float* feats       = T1 + B_SZ * P_SZ;              // 128*2048
  float* name_logits = feats + B_SZ * P_SZ;           // 128*64
  float* ALL         = name_logits + B_SZ * N_NAMES;  // 128*64*128
  float* partials    = ALL + B_SZ * N_NAMES * N_ARGS; // 8*128*2048 (reused)

  // outputs: [loss(1)][scores(B*C)][preds(B)]
  float* out_loss   = (float*)d_out;
  float* out_scores = out_loss + 1;
  float* out_preds  = out_scores + (size_t)B_SZ * C_FULL;

  const long long bp = (long long)B_SZ * P_SZ;        // 262144
  const long long ball = (long long)B_SZ * N_NAMES * N_ARGS;  // 1048576

  // 1) pooled gather
  pool_kernel<<<B_SZ, 256, 0, stream>>>(hidden, amask, pooled);

  // 2) T1 = silu(pooled @ W1)   split-K=8, 32x64 tiles -> 1024 waves
  gemm_wmma_f32<<<dim3(P_SZ / 64, B_SZ / 32, 8), 32, 0, stream>>>(
      pooled, H_SZ, W1, P_SZ, 0, partials, P_SZ, 0, bp, 8, H_SZ / 8);
  reduce_epilogue<<<(int)(bp / 256), 256, 0, stream>>>(
      partials, bp, 8, T1, P_SZ, nullptr, 1, 1, nullptr, nullptr, nullptr, 0);

  // 3) feats = silu(T1 @ W2) + adapter_scale * adapter_emb[adapter_ids]
  gemm_wmma_f32<<<dim3(P_SZ / 64, B_SZ / 32, 8), 32, 0, stream>>>(
      T1, P_SZ, W2, P_SZ, 0, partials, P_SZ, 0, bp, 8, P_SZ / 8);
  reduce_epilogue<<<(int)(bp / 256), 256, 0, stream>>>(
      partials, bp, 8, feats, P_SZ, nullptr, 1, 1,
      adap_emb, adap_ids, adap_sc, P_SZ);

  // 4) name_logits = feats @ Wname + bname   split-K=16 -> 64 waves
  const long long bn = (long long)B_SZ * N_NAMES;     // 8192
  gemm_wmma_f32<<<dim3(N_NAMES / 64, B_SZ / 32, 16), 32, 0, stream>>>(
      feats, P_SZ, Wname, N_NAMES, 0, partials, N_NAMES, 0, bn, 16, P_SZ / 16);
  reduce_epilogue<<<(int)(bn / 256), 256, 0, stream>>>(
      partials, bn, 16, name_logits, N_NAMES, bname, N_NAMES, 0,
      nullptr, nullptr, nullptr, 0);

  // 5) ALL[b,n,a] = feats[b] @ Warg[n][:,a] + barg[n,a]; z=64 names, split-K=2
  gemm_wmma_f32<<<dim3(N_ARGS / 64, B_SZ / 32, N_NAMES * 2), 32, 0, stream>>>(
      feats, P_SZ, Warg, N_ARGS, (long long)P_SZ * N_ARGS,
      partials, N_NAMES * N_ARGS, (long long)N_ARGS, ball, 2, P_SZ / 2);
  reduce_epilogue<<<(int)(ball / 256), 256, 0, stream>>>(
      partials, ball, 2, ALL, N_NAMES * N_ARGS, barg, N_NAMES * N_ARGS, 0,
      nullptr, nullptr, nullptr, 0);

  // 6) loss
  loss_kernel<<<1, B_SZ, 0, stream>>>(name_logits, name_cm, tgt_name,
                                      ALL, arg_cm, tgt_arg, out_loss);

  // 7) scores + preds
  scores_kernel<<<B_SZ, 256, 0, stream>>>(name_logits, ALL, full_cn, full_ca,
                                          full_cm, out_scores, out_preds);
}